// ClearMeshLoss_30176440222506
// MI455X (gfx1250) — compile-verified
//
#include <hip/hip_runtime.h>
#include <hip/hip_bf16.h>

typedef __attribute__((ext_vector_type(2))) float v2f;
typedef __attribute__((ext_vector_type(8))) float v8f;

#define WAVES_PER_BLOCK 8
#define ROWS_PER_WAVE   16
#define ROWS_PER_BLOCK  (WAVES_PER_BLOCK * ROWS_PER_WAVE)  // 128
#define SEG_COLS        1024                               // targets per block (32KB LDS stage)
#define HASH_BITS 18
#define HASH_SIZE (1u << HASH_BITS)
#define HASH_MASK (HASH_SIZE - 1u)
#define EMPTY_KEY 0xFFFFFFFFu
#define BIG_F 3.0e38f

// ---------------------------------------------------------------------------
// order-preserving float <-> uint mapping (total order, works for negatives)
__device__ inline unsigned f32_sortable(float f) {
    unsigned u = __float_as_uint(f);
    return (u & 0x80000000u) ? ~u : (u | 0x80000000u);
}
__device__ inline float f32_unsortable(unsigned s) {
    unsigned u = (s & 0x80000000u) ? (s & 0x7FFFFFFFu) : ~s;
    return __uint_as_float(u);
}

// ---------------------------------------------------------------------------
// block-wide float sum -> atomicAdd(target)
__device__ inline void block_reduce_add(float v, float* target) {
    __shared__ float sm[256];
    sm[threadIdx.x] = v;
    __syncthreads();
    for (int s = blockDim.x >> 1; s > 0; s >>= 1) {
        if ((int)threadIdx.x < s) sm[threadIdx.x] += sm[threadIdx.x + s];
        __syncthreads();
    }
    if (threadIdx.x == 0) atomicAdd(target, sm[0]);
}

// ---------------------------------------------------------------------------
// init accumulators, packed-min arrays, hash table (ws poisoned; re-init per call)
__global__ void init_ws_kernel(float* acc, int* icnt,
                               unsigned long long* pk0, int n0,
                               unsigned long long* pk1, int n1,
                               unsigned* keys, int* cnt, int* f0, int* f1,
                               int tabsize) {
    int i = blockIdx.x * blockDim.x + threadIdx.x;
    if (i < 8) acc[i] = 0.0f;
    if (i < 4) icnt[i] = 0;
    if (i < n0) pk0[i] = ~0ull;
    if (i < n1) pk1[i] = ~0ull;
    if (i < tabsize) {
        keys[i] = EMPTY_KEY;
        cnt[i]  = 0;
        f0[i]   = 0;
        f1[i]   = 0;
    }
}

// ---------------------------------------------------------------------------
// SDF L1: sum |a-b|
__global__ void sdf_abs_kernel(const float* __restrict__ a,
                               const float* __restrict__ b,
                               int n, float* acc) {
    float v = 0.0f;
    int stride = gridDim.x * blockDim.x;
    for (int i = blockIdx.x * blockDim.x + threadIdx.x; i < n; i += stride)
        v += fabsf(a[i] - b[i]);
    block_reduce_add(v, acc);
}

// ---------------------------------------------------------------------------
// Fused NN sweep with V_WMMA_F32_16X16X4_F32 and LDS-staged targets.
// Block = 8 waves x 16 query rows; blockIdx.y selects a 1024-target segment.
// LDS holds two pre-swizzled B images (one per lane group) so the inner loop
// is just ds_load_b128 + wmma + fma + min-update per 8 rows.
// Row mins track (dist - |q|^2); |q|^2 is a per-row constant so ordering is
// preserved; added back before the packed-u64 atomicMin merge.
template <bool NEED_ARG>
__global__ void __launch_bounds__(256)
nn_sweep_kernel(const float* __restrict__ Q, int Nq,
                const float* __restrict__ T, int Nt,
                unsigned long long* __restrict__ packedOut) {
    __shared__ float4 sB[2 * SEG_COLS];   // [grp][col] = (B.x, B.y, |t|^2, 0)

    const int lane    = threadIdx.x & 31;
    const int wave    = threadIdx.x >> 5;
    const int grp     = lane >> 4;        // 0: K={0,1}, 1: K={2,3}
    const int n       = lane & 15;
    const int rowBase = blockIdx.x * ROWS_PER_BLOCK + wave * ROWS_PER_WAVE;
    const int segBase = blockIdx.y * SEG_COLS;

    // ---- cooperative stage: both lane-group operand images
    for (int c = threadIdx.x; c < SEG_COLS; c += blockDim.x) {
        int col = segBase + c;
        int cc  = (col < Nt) ? col : (Nt - 1);
        float t0 = T[cc * 3 + 0];
        float t1 = T[cc * 3 + 1];
        float t2 = T[cc * 3 + 2];
        float g2 = (col < Nt) ? (t0 * t0 + t1 * t1 + t2 * t2) : BIG_F;
        sB[c]            = make_float4(t0, t1, g2, 0.0f);  // grp 0: K={0,1}
        sB[SEG_COLS + c] = make_float4(t2, 0.0f, g2, 0.0f);// grp 1: K={2,3}
    }
    __syncthreads();

    // ---- A tile (16x4 f32, K=3 padded): ISA layout, lane grp picks K pair
    int arow = rowBase + n;  if (arow >= Nq) arow = Nq - 1;
    v2f A;
    A.x = Q[arow * 3 + 2 * grp];
    A.y = grp ? 0.0f : Q[arow * 3 + 1];

    // ---- squared norms of the 8 C/D rows this lane owns (m = r + 8*grp)
    float p2row[8];
#pragma unroll
    for (int r = 0; r < 8; ++r) {
        int rr = rowBase + r + 8 * grp;  if (rr >= Nq) rr = Nq - 1;
        float x = Q[rr * 3 + 0], y = Q[rr * 3 + 1], z = Q[rr * 3 + 2];
        p2row[r] = x * x + y * y + z * z;
    }

    const float4* myB = sB + grp * SEG_COLS + n;

    float bestAdj[8];   // min over cols of (g2 - 2*dot)  ==  min dist - p2row
    int   bestj[8];
#pragma unroll
    for (int r = 0; r < 8; ++r) { bestAdj[r] = BIG_F; bestj[r] = 0; }

#pragma unroll 2
    for (int jt = 0; jt < SEG_COLS / 16; ++jt) {
        float4 t = myB[jt * 16];                    // one ds_load_b128
        v2f B;  B.x = t.x;  B.y = t.y;

        v8f C = {};
        C = __builtin_amdgcn_wmma_f32_16x16x4_f32(
                false, A, false, B, (short)0, C, false, false);

        int col = segBase + jt * 16 + n;
#pragma unroll
        for (int r = 0; r < 8; ++r) {
            float e = fmaf(-2.0f, C[r], t.z);       // dist - p2row (const/row)
            if (NEED_ARG) {
                if (e < bestAdj[r]) { bestAdj[r] = e; bestj[r] = col; }
            } else {
                bestAdj[r] = fminf(bestAdj[r], e);
            }
        }
    }

    // ---- butterfly min(+argmin) across each 16-lane group (same row => same p2)
#pragma unroll
    for (int r = 0; r < 8; ++r) {
        float bv = bestAdj[r]; int bj = bestj[r];
#pragma unroll
        for (int off = 1; off < 16; off <<= 1) {
            float ov = __shfl_xor(bv, off, 32);
            if (NEED_ARG) {
                int oj = __shfl_xor(bj, off, 32);
                if (ov < bv || (ov == bv && oj < bj)) { bv = ov; bj = oj; }
            } else {
                bv = fminf(bv, ov);
            }
        }
        bestAdj[r] = bv; bestj[r] = bj;
    }

    if (n == 0) {   // lanes 0 and 16 own rows rowBase+[0..7]+8*grp
#pragma unroll
        for (int r = 0; r < 8; ++r) {
            int row = rowBase + r + 8 * grp;
            if (row < Nq) {
                float d = bestAdj[r] + p2row[r];
                unsigned long long key =
                    ((unsigned long long)f32_sortable(d) << 32) |
                    (unsigned)(NEED_ARG ? bestj[r] : 0);
                atomicMin(&packedOut[row], key);
            }
        }
    }
}

// ---------------------------------------------------------------------------
// unpack packed (dist,idx) rows: emit argmin indices + sum of min dists
__global__ void unpack_reduce_kernel(const unsigned long long* __restrict__ packed,
                                     int n, int* __restrict__ argOut, float* sumAcc) {
    float v = 0.0f;
    int stride = gridDim.x * blockDim.x;
    for (int i = blockIdx.x * blockDim.x + threadIdx.x; i < n; i += stride) {
        unsigned long long p = packed[i];
        v += f32_unsortable((unsigned)(p >> 32));
        if (argOut) argOut[i] = (int)(unsigned)(p & 0xFFFFFFFFu);
    }
    block_reduce_add(v, sumAcc);
}

// ---------------------------------------------------------------------------
// normal loss: sum (1 - |cos(pred_n_i, gt_n_{argmin_i})|)
__global__ void normal_loss_kernel(const float* __restrict__ pn,
                                   const float* __restrict__ gn,
                                   const int* __restrict__ nn_idx,
                                   int N, float* acc) {
    float v = 0.0f;
    int stride = gridDim.x * blockDim.x;
    for (int i = blockIdx.x * blockDim.x + threadIdx.x; i < N; i += stride) {
        int j = nn_idx[i];
        float px = pn[i * 3 + 0], py = pn[i * 3 + 1], pz = pn[i * 3 + 2];
        float gx = gn[j * 3 + 0], gy = gn[j * 3 + 1], gz = gn[j * 3 + 2];
        float lp = fmaxf(sqrtf(px * px + py * py + pz * pz), 1e-8f);
        float lg = fmaxf(sqrtf(gx * gx + gy * gy + gz * gz), 1e-8f);
        float c  = (px * gx + py * gy + pz * gz) / (lp * lg);
        v += 1.0f - fabsf(c);
    }
    block_reduce_add(v, acc);
}

// ---------------------------------------------------------------------------
// face normals (normalized with EPS_NRM = 1e-12)
__global__ void face_normals_kernel(const float* __restrict__ V,
                                    const int* __restrict__ F,
                                    int nf, float* __restrict__ out) {
    int f = blockIdx.x * blockDim.x + threadIdx.x;
    if (f >= nf) return;
    int i0 = F[f * 3 + 0], i1 = F[f * 3 + 1], i2 = F[f * 3 + 2];
    float ax = V[i1 * 3 + 0] - V[i0 * 3 + 0];
    float ay = V[i1 * 3 + 1] - V[i0 * 3 + 1];
    float az = V[i1 * 3 + 2] - V[i0 * 3 + 2];
    float bx = V[i2 * 3 + 0] - V[i0 * 3 + 0];
    float by = V[i2 * 3 + 1] - V[i0 * 3 + 1];
    float bz = V[i2 * 3 + 2] - V[i0 * 3 + 2];
    float nx = ay * bz - az * by;
    float ny = az * bx - ax * bz;
    float nz = ax * by - ay * bx;
    float l = fmaxf(sqrtf(nx * nx + ny * ny + nz * nz), 1e-12f);
    out[f * 3 + 0] = nx / l;
    out[f * 3 + 1] = ny / l;
    out[f * 3 + 2] = nz / l;
}

// ---------------------------------------------------------------------------
// edge hash insert: key = lo*nv + hi, linear probing, capture first 2 faces
__global__ void edge_insert_kernel(const int* __restrict__ F, int nf, int nv,
                                   unsigned* keys, int* cnt, int* f0, int* f1) {
    int f = blockIdx.x * blockDim.x + threadIdx.x;
    if (f >= nf) return;
    int v0 = F[f * 3 + 0], v1 = F[f * 3 + 1], v2 = F[f * 3 + 2];
    int ea[3] = { v0, v1, v2 };
    int eb[3] = { v1, v2, v0 };
#pragma unroll
    for (int e = 0; e < 3; ++e) {
        int lo = min(ea[e], eb[e]);
        int hi = max(ea[e], eb[e]);
        unsigned key  = (unsigned)lo * (unsigned)nv + (unsigned)hi;
        unsigned slot = ((key * 2654435761u) >> (32 - HASH_BITS)) & HASH_MASK;
        for (;;) {
            unsigned prev = atomicCAS(&keys[slot], EMPTY_KEY, key);
            if (prev == EMPTY_KEY || prev == key) {
                int arrival = atomicAdd(&cnt[slot], 1);
                if (arrival == 0)      f0[slot] = f;
                else if (arrival == 1) f1[slot] = f;
                break;
            }
            slot = (slot + 1) & HASH_MASK;
        }
    }
}

// ---------------------------------------------------------------------------
// walk hash table: unique-edge count, pair count, relu(cos - 0.5) sum
__global__ void edge_accum_kernel(const unsigned* __restrict__ keys,
                                  const int* __restrict__ cnt,
                                  const int* __restrict__ f0,
                                  const int* __restrict__ f1,
                                  const float* __restrict__ fn,
                                  int tabsize, float* edgeAcc, int* icnt) {
    int s = blockIdx.x * blockDim.x + threadIdx.x;
    if (s >= tabsize) return;
    if (keys[s] == EMPTY_KEY) return;
    atomicAdd(&icnt[0], 1);
    if (cnt[s] == 2) {
        atomicAdd(&icnt[1], 1);
        int a = f0[s], b = f1[s];
        float c = fn[a * 3 + 0] * fn[b * 3 + 0]
                + fn[a * 3 + 1] * fn[b * 3 + 1]
                + fn[a * 3 + 2] * fn[b * 3 + 2];
        float t = fmaxf(c - 0.5f, 0.0f);
        atomicAdd(edgeAcc, t);
    }
}

// ---------------------------------------------------------------------------
// combine scalars (weights: chamfer 1.0, normal 0.5, edge 0.3, wt 0.2, sdf 1.0)
__global__ void finalize_kernel(const float* acc, const int* icnt,
                                int NS, int N, int M, float* out) {
    if (threadIdx.x != 0 || blockIdx.x != 0) return;
    float sdf_l     = acc[0] / (float)NS;
    float chamfer_l = acc[1] / (float)N + acc[2] / (float)M;
    float normal_l  = 0.5f * (acc[3] / (float)N);
    int uniq = icnt[0], pairs = icnt[1];
    float edge_l = 0.3f * (pairs > 0 ? acc[4] / (float)pairs : 0.0f);
    float wt_l   = 0.2f * (uniq  > 0 ? (float)(uniq - pairs) / (float)uniq : 0.0f);
    out[0] = sdf_l;
    out[1] = chamfer_l;
    out[2] = normal_l;
    out[3] = edge_l;
    out[4] = wt_l;
    out[5] = sdf_l + chamfer_l + normal_l + edge_l + wt_l;
}

// ---------------------------------------------------------------------------
extern "C" void kernel_launch(void* const* d_in, const int* in_sizes, int n_in,
                              void* d_out, int out_size, void* d_ws, size_t ws_size,
                              hipStream_t stream) {
    const float* pred_sdf = (const float*)d_in[0];
    const float* gt_sdf   = (const float*)d_in[1];
    const float* ex_verts = (const float*)d_in[2];
    const int*   ex_faces = (const int*)  d_in[3];
    // d_in[4], d_in[5] (gt mesh) unused by the reference losses
    const float* pred_pts = (const float*)d_in[6];
    const float* gt_pts   = (const float*)d_in[7];
    const float* pred_nrm = (const float*)d_in[8];
    const float* gt_nrm   = (const float*)d_in[9];

    const int NS = in_sizes[0];          // 65536
    const int V  = in_sizes[2] / 3;      // 20000
    const int F  = in_sizes[3] / 3;      // 40000
    const int N  = in_sizes[6] / 3;      // 8192 pred points
    const int M  = in_sizes[7] / 3;      // 8192 gt points

    // ---- workspace carve-up (256B aligned chunks)
    char* wsp = (char*)d_ws;
    auto carve = [&](size_t bytes) {
        void* p = (void*)wsp;
        wsp += (bytes + 255) & ~(size_t)255;
        return p;
    };
    float* acc  = (float*)carve(8 * sizeof(float));  // [0]sdf [1]p2g [2]g2p [3]normal [4]edge
    int*   icnt = (int*)  carve(4 * sizeof(int));    // [0]unique [1]pairs
    unsigned long long* pkP2G = (unsigned long long*)carve((size_t)N * sizeof(unsigned long long));
    unsigned long long* pkG2P = (unsigned long long*)carve((size_t)M * sizeof(unsigned long long));
    int*      argP2G = (int*)     carve((size_t)N * sizeof(int));
    float*    fnorm  = (float*)   carve((size_t)F * 3 * sizeof(float));
    unsigned* hkeys  = (unsigned*)carve((size_t)HASH_SIZE * sizeof(unsigned));
    int*      hcnt   = (int*)     carve((size_t)HASH_SIZE * sizeof(int));
    int*      hf0    = (int*)     carve((size_t)HASH_SIZE * sizeof(int));
    int*      hf1    = (int*)     carve((size_t)HASH_SIZE * sizeof(int));

    // 1) init accumulators + packed-min arrays + hash table
    init_ws_kernel<<<(HASH_SIZE + 255) / 256, 256, 0, stream>>>(
        acc, icnt, pkP2G, N, pkG2P, M, hkeys, hcnt, hf0, hf1, (int)HASH_SIZE);

    // 2) SDF L1 sum
    sdf_abs_kernel<<<256, 256, 0, stream>>>(pred_sdf, gt_sdf, NS, &acc[0]);

    // 3) chamfer pred->gt with argmin (WMMA sweep, segmented targets)
    {
        dim3 grid((N + ROWS_PER_BLOCK - 1) / ROWS_PER_BLOCK,
                  (M + SEG_COLS - 1) / SEG_COLS);
        nn_sweep_kernel<true><<<grid, 256, 0, stream>>>(pred_pts, N, gt_pts, M, pkP2G);
    }

    // 4) chamfer gt->pred (no argmin needed)
    {
        dim3 grid((M + ROWS_PER_BLOCK - 1) / ROWS_PER_BLOCK,
                  (N + SEG_COLS - 1) / SEG_COLS);
        nn_sweep_kernel<false><<<grid, 256, 0, stream>>>(gt_pts, M, pred_pts, N, pkG2P);
    }

    // 5) unpack mins: chamfer sums + argmin indices
    unpack_reduce_kernel<<<32, 256, 0, stream>>>(pkP2G, N, argP2G, &acc[1]);
    unpack_reduce_kernel<<<32, 256, 0, stream>>>(pkG2P, M, (int*)nullptr, &acc[2]);

    // 6) normal loss using argmin indices
    normal_loss_kernel<<<64, 256, 0, stream>>>(pred_nrm, gt_nrm, argP2G, N, &acc[3]);

    // 7) face normals of extracted mesh
    face_normals_kernel<<<(F + 255) / 256, 256, 0, stream>>>(ex_verts, ex_faces, F, fnorm);

    // 8) edge hash insert (sort-free run-length detection)
    edge_insert_kernel<<<(F + 255) / 256, 256, 0, stream>>>(
        ex_faces, F, V, hkeys, hcnt, hf0, hf1);

    // 9) edge sharpness + watertight accumulation
    edge_accum_kernel<<<(HASH_SIZE + 255) / 256, 256, 0, stream>>>(
        hkeys, hcnt, hf0, hf1, fnorm, (int)HASH_SIZE, &acc[4], icnt);

    // 10) combine the six scalar losses
    finalize_kernel<<<1, 1, 0, stream>>>(acc, icnt, NS, N, M, (float*)d_out);
}